// AtlasMAGBlock_5136780886737
// MI455X (gfx1250) — compile-verified
//
#include <hip/hip_runtime.h>
#include <math.h>
#include <stdint.h>

// Problem constants (from the reference)
#define DD    256
#define BB    2
#define LL    256
#define HH    16
#define HDIM  16
#define SWIN  128
#define NPERS 4
#define OMEGA 8
#define BL    (BB*LL)

typedef __attribute__((ext_vector_type(2))) float v2f;
typedef __attribute__((ext_vector_type(8))) float v8f;
typedef __attribute__((ext_vector_type(4))) unsigned int u32x4;
typedef __attribute__((ext_vector_type(8))) int i32x8;
typedef __attribute__((ext_vector_type(4))) int i32x4;

#if __has_builtin(__builtin_amdgcn_tensor_load_to_lds) && __has_builtin(__builtin_amdgcn_s_wait_tensorcnt)
#define HAVE_TDM 1
#else
#define HAVE_TDM 0
#endif

// D = A(16x4) * B(4x16) + C, fp32, wave32.
__device__ __forceinline__ v8f wmma4(v2f a, v2f b, v8f c) {
  return __builtin_amdgcn_wmma_f32_16x16x4_f32(false, a, false, b, (short)0, c, false, false);
}

__device__ __forceinline__ float sigmoidf_(float x) { return 1.f / (1.f + __expf(-x)); }

// ---------------- LayerNorm: one block (256 thr) per row ----------------
__global__ void __launch_bounds__(256)
k_ln(const float* __restrict__ x, const float* __restrict__ w,
     const float* __restrict__ b, float* __restrict__ o) {
  __shared__ float red[DD];
  const int row = blockIdx.x, e = threadIdx.x;
  const float v = x[(size_t)row*DD + e];
  red[e] = v; __syncthreads();
  for (int s = 128; s > 0; s >>= 1) { if (e < s) red[e] += red[e+s]; __syncthreads(); }
  const float mean = red[0] * (1.f/DD);
  __syncthreads();
  const float d0 = v - mean;
  red[e] = d0*d0; __syncthreads();
  for (int s = 128; s > 0; s >>= 1) { if (e < s) red[e] += red[e+s]; __syncthreads(); }
  const float var = red[0] * (1.f/DD);
  o[(size_t)row*DD + e] = d0 * rsqrtf(var + 1e-5f) * w[e] + b[e];
}

// ---------------- Row L2-normalize (in place), block per row ----------------
__global__ void __launch_bounds__(256)
k_rownorm(float* __restrict__ z) {
  __shared__ float red[DD];
  const int row = blockIdx.x, e = threadIdx.x;
  const float vv = z[(size_t)row*DD + e];
  red[e] = vv*vv; __syncthreads();
  for (int s = 128; s > 0; s >>= 1) { if (e < s) red[e] += red[e+s]; __syncthreads(); }
  const float n = fmaxf(sqrtf(red[0]), 1e-12f);
  z[(size_t)row*DD + e] = vv / n;
}

// ---------------- zcat = [z, z*z] for the poly features ----------------
__global__ void k_zcat(const float* __restrict__ z, float* __restrict__ o) {
  const int i = blockIdx.x * blockDim.x + threadIdx.x;
  if (i >= BL * 2 * DD) return;
  const int r = i >> 9, j = i & 511;
  const float vv = z[(size_t)r*DD + (j & 255)];
  o[i] = (j < 256) ? vv : vv * vv;
}

// ---------------- gamma gate: sigmoid(xn @ gamma_w + gb), N=1 ----------------
__global__ void __launch_bounds__(256)
k_gamma(const float* __restrict__ xn, const float* __restrict__ gw,
        const float* __restrict__ gb, float* __restrict__ g) {
  __shared__ float red[DD];
  const int row = blockIdx.x, e = threadIdx.x;
  red[e] = xn[(size_t)row*DD + e] * gw[e];
  __syncthreads();
  for (int s = 128; s > 0; s >>= 1) { if (e < s) red[e] += red[e+s]; __syncthreads(); }
  if (e == 0) g[row] = sigmoidf_(red[0] + gb[0]);
}

// ---------------- broadcast M_init/S_init into per-batch state ----------------
__global__ void k_init_state(const float* __restrict__ Mi, const float* __restrict__ Si,
                             float* __restrict__ Mg, float* __restrict__ Sg) {
  const int i = blockIdx.x * blockDim.x + threadIdx.x;
  if (i >= BB*DD*DD) return;
  const int p = i & (DD*DD - 1);
  Mg[i] = Mi[p];
  Sg[i] = Si[p];
}

// ---------------- WMMA GEMM with TDM-staged W column slice ----------------
// Block = 8 waves covering 128 rows x 16 cols; W[:, n0:n0+16] DMA'd to LDS.
// A [M,K] row-major, W [K,N] row-major. act: 0 none, 1 sigmoid, 2 sigmoid*0.1
#define GEMM_MAX_K 512

__global__ void __launch_bounds__(256)
k_gemm(const float* __restrict__ A, const float* __restrict__ W,
       const float* __restrict__ bias, const float* __restrict__ resid,
       float* __restrict__ C, int M, int N, int K, int act) {
  __shared__ float sW[GEMM_MAX_K * 16];   // [k][16] column slice of W

  const int tilesN = N >> 4;
  const int nb = blockIdx.x % tilesN;
  const int mb = blockIdx.x / tilesN;
  const int n0 = nb << 4;
  const int wave = threadIdx.x >> 5;
  const int lane = threadIdx.x & 31;
  const int lm = lane & 15;
  const int ko = (lane >> 4) << 1;
  const int m0 = ((mb << 3) + wave) << 4;

#if HAVE_TDM
  // Tensor Data Mover: DMA the [K x 16] column slice of W into LDS.
  if (threadIdx.x < 32) {
    const uint64_t gaddr = (uint64_t)(uintptr_t)(W + n0);
    const uint32_t ldsoff = (uint32_t)(uintptr_t)(&sW[0]);
    u32x4 g0;
    g0.x = 1u;                                   // count=1 (valid user descriptor)
    g0.y = ldsoff;                               // lds_addr (bytes)
    g0.z = (uint32_t)gaddr;                      // global_addr[31:0]
    g0.w = (uint32_t)(gaddr >> 32) | (2u << 30); // global_addr[56:32] | type=2
    i32x8 g1;
    g1[0] = (int)(2u << 16);                     // data_size=4B; wg_mask=0
    g1[1] = (int)(16u << 16);                    // tensor_dim0 = 16 (lo16)
    g1[2] = (int)(((uint32_t)K & 0xFFFFu) << 16);// tensor_dim1 = K (lo16)
    g1[3] = (int)(16u << 16);                    // tile_dim0 = 16
    g1[4] = K;                                   // tile_dim1 = K, tile_dim2 = 0
    g1[5] = N;                                   // tensor_dim0_stride = N
    g1[6] = 0;
    g1[7] = 0;
    const i32x4 z4 = {0, 0, 0, 0};               // groups 2/3 unused (2D tensor)
    const i32x8 z8 = {0, 0, 0, 0, 0, 0, 0, 0};
    __builtin_amdgcn_tensor_load_to_lds(g0, g1, z4, z4, z8, 0);
    __builtin_amdgcn_s_wait_tensorcnt((short)0);
  }
#else
  for (int i = threadIdx.x; i < K * 16; i += 256) {
    const int kk = i >> 4, c = i & 15;
    sW[i] = W[(size_t)kk * N + n0 + c];
  }
#endif
  __syncthreads();

  const float* Arow = A + (size_t)(m0 + lm) * K;
  v8f acc = {0.f,0.f,0.f,0.f,0.f,0.f,0.f,0.f};
  #pragma unroll 4
  for (int k0 = 0; k0 < K; k0 += 4) {
    if ((k0 & 63) == 0 && k0 + 64 < K) __builtin_prefetch(Arow + k0 + 64, 0, 1);
    v2f a, b;
    a.x = Arow[k0 + ko];
    a.y = Arow[k0 + ko + 1];
    b.x = sW[(k0 + ko) * 16 + lm];
    b.y = sW[(k0 + ko + 1) * 16 + lm];
    acc = wmma4(a, b, acc);
  }
  const int rbase = m0 + ((lane >> 4) << 3);
  #pragma unroll
  for (int vg = 0; vg < 8; ++vg) {
    const int r = rbase + vg, cc = n0 + lm;
    float v = acc[vg];
    if (bias) v += bias[cc];
    if (act == 1) v = sigmoidf_(v);
    else if (act == 2) v = sigmoidf_(v) * 0.1f;
    if (resid) v += resid[(size_t)r * N + cc];
    C[(size_t)r * N + cc] = v;
  }
}

// ---------------- Sliding-window attention w/ persistent tokens ----------------
// One thread per (b, l, h); online softmax over [4 persistent + <=128 local] keys.
__global__ void __launch_bounds__(256)
k_attn(const float* __restrict__ q, const float* __restrict__ k,
       const float* __restrict__ v, const float* __restrict__ pk,
       const float* __restrict__ pv, float* __restrict__ yh) {
  const int t = blockIdx.x * blockDim.x + threadIdx.x;
  if (t >= BB*LL*HH) return;
  const int h = t & (HH-1);
  const int l = (t >> 4) & (LL-1);
  const int b = t >> 12;
  const int col = h * HDIM;
  const float* qp = q + (size_t)(b*LL + l)*DD + col;
  float qv[HDIM];
  #pragma unroll
  for (int j = 0; j < HDIM; ++j) qv[j] = qp[j];
  float mx = -1e30f, ssum = 0.f;
  float accv[HDIM];
  #pragma unroll
  for (int j = 0; j < HDIM; ++j) accv[j] = 0.f;

  auto proc = [&](const float* kp2, const float* vp2) {
    float s = 0.f;
    #pragma unroll
    for (int j = 0; j < HDIM; ++j) s += qv[j] * kp2[j];
    s *= 0.25f;                      // 1/sqrt(16)
    const float nmx = fmaxf(mx, s);
    const float sc = __expf(mx - nmx);
    const float w  = __expf(s - nmx);
    ssum = ssum * sc + w;
    #pragma unroll
    for (int j = 0; j < HDIM; ++j) accv[j] = accv[j] * sc + w * vp2[j];
    mx = nmx;
  };

  for (int p = 0; p < NPERS; ++p)
    proc(pk + (size_t)p*DD + col, pv + (size_t)p*DD + col);
  int m0 = l - (SWIN - 1); if (m0 < 0) m0 = 0;
  for (int m = m0; m <= l; ++m)
    proc(k + (size_t)(b*LL + m)*DD + col, v + (size_t)(b*LL + m)*DD + col);

  const float inv = 1.f / ssum;
  float* op = yh + (size_t)(b*LL + l)*DD + col;
  #pragma unroll
  for (int j = 0; j < HDIM; ++j) op[j] = accv[j] * inv;
}

// ---------------- MAG gate: g = sigmoid(ln(ya)*ga) * (ln(ym)*gm) ----------------
__global__ void __launch_bounds__(256)
k_gate(const float* __restrict__ ya_in, const float* __restrict__ ym_in,
       const float* __restrict__ law, const float* __restrict__ lab,
       const float* __restrict__ lmw, const float* __restrict__ lmb,
       const float* __restrict__ ga, const float* __restrict__ gm,
       float* __restrict__ g) {
  __shared__ float red[DD];
  const int row = blockIdx.x, e = threadIdx.x;
  const float a = ya_in[(size_t)row*DD + e];
  const float m = ym_in[(size_t)row*DD + e];

  red[e] = a; __syncthreads();
  for (int s = 128; s > 0; s >>= 1) { if (e < s) red[e] += red[e+s]; __syncthreads(); }
  const float ma = red[0] * (1.f/DD); __syncthreads();
  const float da = a - ma;
  red[e] = da*da; __syncthreads();
  for (int s = 128; s > 0; s >>= 1) { if (e < s) red[e] += red[e+s]; __syncthreads(); }
  const float na = da * rsqrtf(red[0]*(1.f/DD) + 1e-5f) * law[e] + lab[e];
  __syncthreads();

  red[e] = m; __syncthreads();
  for (int s = 128; s > 0; s >>= 1) { if (e < s) red[e] += red[e+s]; __syncthreads(); }
  const float mm = red[0] * (1.f/DD); __syncthreads();
  const float dm = m - mm;
  red[e] = dm*dm; __syncthreads();
  for (int s = 128; s > 0; s >>= 1) { if (e < s) red[e] += red[e+s]; __syncthreads(); }
  const float nm = dm * rsqrtf(red[0]*(1.f/DD) + 1e-5f) * lmw[e] + lmb[e];

  g[(size_t)row*DD + e] = sigmoidf_(na * ga[e]) * (nm * gm[e]);
}

// ---------------- Atlas memory scan: persistent WG per batch ----------------
// X lives in LDS (padded rows) and feeds the WMMA Newton-Schulz matmuls.
#define LDX 268   // 268%64=12 and (2*268)%64=24 -> conflict-free fragment reads
#define SCAN_SMEM_FLOATS (LDX*DD + 3*OMEGA*DD + 5*DD + OMEGA + 32 + 4)

__global__ void __launch_bounds__(1024)
k_scan(const float* __restrict__ k_phi, const float* __restrict__ vm,
       const float* __restrict__ gam, const float* __restrict__ q_phi,
       const float* __restrict__ alp, const float* __restrict__ eta,
       const float* __restrict__ tht,
       float* __restrict__ Mg, float* __restrict__ Sg,
       float* __restrict__ Xg, float* __restrict__ Tg,
       float* __restrict__ ys) {
  extern __shared__ float sm[];
  float* sX    = sm;                    // LDX*DD
  float* sKw   = sX + LDX*DD;           // 8*256
  float* sVw   = sKw + OMEGA*DD;        // 8*256
  float* sErr  = sVw + OMEGA*DD;        // 8*256
  float* sQ    = sErr + OMEGA*DD;       // 256
  float* sAl   = sQ + DD;               // 256
  float* sEt   = sAl + DD;              // 256
  float* sTh   = sEt + DD;              // 256
  float* sY    = sTh + DD;              // 256
  float* sWgt  = sY + DD;               // 8
  float* sRed  = sWgt + OMEGA;          // 32
  float* sScal = sRed + 32;             // [0]=1/fro, [1]=1/c

  const int b = blockIdx.x;
  const int tid = threadIdx.x;
  const int wave = tid >> 5, lane = tid & 31;
  const int lm = lane & 15, ko = (lane >> 4) << 1;

  float* M = Mg + (size_t)b*DD*DD;
  float* S = Sg + (size_t)b*DD*DD;
  float* X = Xg + (size_t)b*DD*DD;
  float* T = Tg + (size_t)b*DD*DD;

  for (int t = 0; t < LL; ++t) {
    // Phase A: stage window + per-step vectors in LDS
    if (tid < DD) {
      const size_t base = (size_t)(b*LL + t)*DD + tid;
      sQ[tid]  = q_phi[base];
      sAl[tid] = alp[base];
      sEt[tid] = eta[base];
      sTh[tid] = tht[base];
      sY[tid]  = 0.f;
    }
    for (int p = tid; p < OMEGA*DD; p += 1024) {
      const int c = p >> 8, e = p & 255;
      const int src = t - (OMEGA-1) + c;
      const bool ok = (src >= 0);
      sKw[p] = ok ? k_phi[(size_t)(b*LL + src)*DD + e] : 0.f;
      sVw[p] = ok ? vm[(size_t)(b*LL + src)*DD + e] : 0.f;
    }
    if (tid < OMEGA) {
      const int src = t - (OMEGA-1) + tid;
      sWgt[tid] = (src >= 0) ? gam[b*LL + src] : 0.f;
    }
    if (tid == 0) {
      const int c = (t + 1 < OMEGA) ? (t + 1) : OMEGA;
      sScal[1] = 1.f / (float)c;
    }
    __syncthreads();

    // Phase B: err[c,d] = sum_e M[d,e]*kw[c,e] - vw[c,d]; wave handles 8 d's
    for (int dd = wave*8; dd < wave*8 + 8; ++dd) {
      float acc[OMEGA];
      #pragma unroll
      for (int c = 0; c < OMEGA; ++c) acc[c] = 0.f;
      const float* Mrow = M + (size_t)dd*DD;
      for (int e = lane; e < DD; e += 32) {
        const float mv = Mrow[e];
        #pragma unroll
        for (int c = 0; c < OMEGA; ++c) acc[c] += mv * sKw[c*DD + e];
      }
      #pragma unroll
      for (int c = 0; c < OMEGA; ++c)
        for (int off = 16; off; off >>= 1) acc[c] += __shfl_xor(acc[c], off, 32);
      if (lane == 0) {
        #pragma unroll
        for (int c = 0; c < OMEGA; ++c) sErr[c*DD + dd] = acc[c] - sVw[c*DD + dd];
      }
    }
    __syncthreads();

    // Phase C: grad (rank-8) + S update + Frobenius partial
    float fsum = 0.f;
    const float invC = sScal[1];
    for (int p = tid; p < DD*DD; p += 1024) {
      const int d = p >> 8, e = p & 255;
      float gsum = 0.f;
      #pragma unroll
      for (int c = 0; c < OMEGA; ++c)
        gsum += sWgt[c] * sErr[c*DD + d] * sKw[c*DD + e];
      const float sv = sTh[d] * S[p] + gsum * invC;
      S[p] = sv;
      fsum += sv * sv;
    }
    for (int off = 16; off; off >>= 1) fsum += __shfl_xor(fsum, off, 32);
    if (lane == 0) sRed[wave] = fsum;
    __syncthreads();
    if (tid == 0) {
      float s = 0.f;
      for (int w = 0; w < 32; ++w) s += sRed[w];
      sScal[0] = 1.f / (sqrtf(s) + 1e-7f);
    }
    __syncthreads();

    // Phase C2: X = S / fro into padded LDS
    {
      const float rf = sScal[0];
      for (int p = tid; p < DD*DD; p += 1024) {
        const int d = p >> 8, e = p & 255;
        sX[d*LDX + e] = S[p] * rf;
      }
    }
    __syncthreads();

    // Phase D: T = X * X^T  (WMMA f32 16x16x4, A & B fragments from LDS)
    for (int tt = wave*8; tt < wave*8 + 8; ++tt) {
      const int m0 = (tt >> 4) << 4, n0 = (tt & 15) << 4;
      v8f acc = {0.f,0.f,0.f,0.f,0.f,0.f,0.f,0.f};
      for (int k0 = 0; k0 < DD; k0 += 4) {
        v2f a, bb;
        a.x  = sX[(m0 + lm)*LDX + k0 + ko];
        a.y  = sX[(m0 + lm)*LDX + k0 + ko + 1];
        bb.x = sX[(n0 + lm)*LDX + k0 + ko];     // B[k,n] = X[n,k]
        bb.y = sX[(n0 + lm)*LDX + k0 + ko + 1];
        acc = wmma4(a, bb, acc);
      }
      const int rbase = m0 + ((lane >> 4) << 3);
      #pragma unroll
      for (int vg = 0; vg < 8; ++vg)
        T[(size_t)(rbase + vg)*DD + n0 + lm] = acc[vg];
    }
    __threadfence();
    __syncthreads();

    // Phase E: newX = 1.5*X - 0.5*(T @ X)  -> global X buffer
    for (int tt = wave*8; tt < wave*8 + 8; ++tt) {
      const int m0 = (tt >> 4) << 4, n0 = (tt & 15) << 4;
      v8f acc = {0.f,0.f,0.f,0.f,0.f,0.f,0.f,0.f};
      for (int k0 = 0; k0 < DD; k0 += 4) {
        v2f a, bb;
        a.x  = T[(size_t)(m0 + lm)*DD + k0 + ko];
        a.y  = T[(size_t)(m0 + lm)*DD + k0 + ko + 1];
        bb.x = sX[(k0 + ko)*LDX + n0 + lm];
        bb.y = sX[(k0 + ko + 1)*LDX + n0 + lm];
        acc = wmma4(a, bb, acc);
      }
      const int rbase = m0 + ((lane >> 4) << 3);
      #pragma unroll
      for (int vg = 0; vg < 8; ++vg) {
        const int r = rbase + vg, cc = n0 + lm;
        X[(size_t)r*DD + cc] = 1.5f * sX[r*LDX + cc] - 0.5f * acc[vg];
      }
    }
    __threadfence();
    __syncthreads();

    // Phase F: M = alpha*M - eta*X ; y = M @ q (wave reduce + ds atomic)
    {
      const int e = tid & 255;
      const float qe = sQ[e];
      for (int p = tid; p < DD*DD; p += 1024) {
        const int d = p >> 8;
        const float mnew = sAl[d] * M[p] - sEt[d] * X[p];
        M[p] = mnew;
        float prod = mnew * qe;
        for (int off = 16; off; off >>= 1) prod += __shfl_xor(prod, off, 32);
        if (lane == 0) atomicAdd(&sY[d], prod);
      }
    }
    __threadfence();
    __syncthreads();
    if (tid < DD) ys[(size_t)(b*LL + t)*DD + tid] = sY[tid];
    __syncthreads();
  }
}

// ---------------- host ----------------
static inline void launch_gemm(const float* A, const float* W, const float* bias,
                               const float* resid, float* C, int M, int N, int K,
                               int act, hipStream_t s) {
  const int blocks = (M >> 7) * (N >> 4);   // 8 waves x 16 rows = 128 rows/block
  k_gemm<<<blocks, 256, 0, s>>>(A, W, bias, resid, C, M, N, K, act);
}

extern "C" void kernel_launch(void* const* d_in, const int* in_sizes, int n_in,
                              void* d_out, int out_size, void* d_ws, size_t ws_size,
                              hipStream_t stream) {
  (void)in_sizes; (void)n_in; (void)out_size; (void)ws_size;
  const float* x        = (const float*)d_in[0];
  const float* ln_pre_w = (const float*)d_in[1];
  const float* ln_pre_b = (const float*)d_in[2];
  const float* attn_wq  = (const float*)d_in[3];
  const float* attn_bq  = (const float*)d_in[4];
  const float* attn_wk  = (const float*)d_in[5];
  const float* attn_bk  = (const float*)d_in[6];
  const float* attn_wv  = (const float*)d_in[7];
  const float* attn_bv  = (const float*)d_in[8];
  const float* attn_wo  = (const float*)d_in[9];
  const float* attn_bo  = (const float*)d_in[10];
  const float* pers_k   = (const float*)d_in[11];
  const float* pers_v   = (const float*)d_in[12];
  const float* mem_wq   = (const float*)d_in[13];
  const float* mem_wk   = (const float*)d_in[14];
  const float* mem_wv   = (const float*)d_in[15];
  const float* P_init   = (const float*)d_in[16];
  const float* poly_w   = (const float*)d_in[17];
  const float* poly_b   = (const float*)d_in[18];
  const float* M_init   = (const float*)d_in[19];
  const float* S_init   = (const float*)d_in[20];
  const float* alpha_w  = (const float*)d_in[21];
  const float* alpha_b  = (const float*)d_in[22];
  const float* eta_w    = (const float*)d_in[23];
  const float* eta_b    = (const float*)d_in[24];
  const float* theta_w  = (const float*)d_in[25];
  const float* theta_b  = (const float*)d_in[26];
  const float* gamma_w  = (const float*)d_in[27];
  const float* gamma_b  = (const float*)d_in[28];
  const float* memout_w = (const float*)d_in[29];
  const float* memout_b = (const float*)d_in[30];
  const float* ln_attn_w= (const float*)d_in[31];
  const float* ln_attn_b= (const float*)d_in[32];
  const float* ln_mem_w = (const float*)d_in[33];
  const float* ln_mem_b = (const float*)d_in[34];
  const float* gamma_at = (const float*)d_in[35];
  const float* gamma_me = (const float*)d_in[36];
  const float* out_w    = (const float*)d_in[37];
  const float* out_b    = (const float*)d_in[38];

  float* ws = (float*)d_ws;
  size_t off = 0;
  auto take = [&](size_t n) { float* p = ws + off; off += n; return p; };
  float* xn    = take((size_t)BL*DD);
  float* qb    = take((size_t)BL*DD);
  float* kb    = take((size_t)BL*DD);
  float* vb    = take((size_t)BL*DD);
  float* yh    = take((size_t)BL*DD);
  float* y_att = take((size_t)BL*DD);
  float* qm    = take((size_t)BL*DD);
  float* kmn   = take((size_t)BL*DD);
  float* vmb   = take((size_t)BL*DD);
  float* q_al  = take((size_t)BL*DD);
  float* kcat  = take((size_t)BL*2*DD);
  float* qcat  = take((size_t)BL*2*DD);
  float* k_phi = take((size_t)BL*DD);
  float* q_phi = take((size_t)BL*DD);
  float* alpha = take((size_t)BL*DD);
  float* etab  = take((size_t)BL*DD);
  float* theta = take((size_t)BL*DD);
  float* gamv  = take((size_t)BL);
  float* Mg    = take((size_t)BB*DD*DD);
  float* Sg    = take((size_t)BB*DD*DD);
  float* Xg    = take((size_t)BB*DD*DD);
  float* Tg    = take((size_t)BB*DD*DD);
  float* ysb   = take((size_t)BL*DD);
  float* y_mem = take((size_t)BL*DD);
  float* gbuf  = take((size_t)BL*DD);

  // pre-LN
  k_ln<<<BL, 256, 0, stream>>>(x, ln_pre_w, ln_pre_b, xn);

  // Branch 1: sliding-window attention
  launch_gemm(xn, attn_wq, attn_bq, nullptr, qb, BL, DD, DD, 0, stream);
  launch_gemm(xn, attn_wk, attn_bk, nullptr, kb, BL, DD, DD, 0, stream);
  launch_gemm(xn, attn_wv, attn_bv, nullptr, vb, BL, DD, DD, 0, stream);
  k_attn<<<(BB*LL*HH + 255)/256, 256, 0, stream>>>(qb, kb, vb, pers_k, pers_v, yh);
  launch_gemm(yh, attn_wo, attn_bo, nullptr, y_att, BL, DD, DD, 0, stream);

  // Branch 2: Atlas memory — feature pipeline
  launch_gemm(xn, mem_wq, nullptr, nullptr, qm,  BL, DD, DD, 0, stream);
  launch_gemm(xn, mem_wk, nullptr, nullptr, kmn, BL, DD, DD, 0, stream);
  launch_gemm(xn, mem_wv, nullptr, nullptr, vmb, BL, DD, DD, 0, stream);
  k_rownorm<<<BL, 256, 0, stream>>>(kmn);
  launch_gemm(qm, P_init, nullptr, nullptr, q_al, BL, DD, DD, 0, stream);
  k_zcat<<<(BL*2*DD + 255)/256, 256, 0, stream>>>(kmn, kcat);
  k_zcat<<<(BL*2*DD + 255)/256, 256, 0, stream>>>(q_al, qcat);
  launch_gemm(kcat, poly_w, poly_b, nullptr, k_phi, BL, DD, 2*DD, 0, stream);
  launch_gemm(qcat, poly_w, poly_b, nullptr, q_phi, BL, DD, 2*DD, 0, stream);

  // gates
  launch_gemm(xn, alpha_w, alpha_b, nullptr, alpha, BL, DD, DD, 1, stream);
  launch_gemm(xn, eta_w,   eta_b,   nullptr, etab,  BL, DD, DD, 2, stream);
  launch_gemm(xn, theta_w, theta_b, nullptr, theta, BL, DD, DD, 1, stream);
  k_gamma<<<BL, 256, 0, stream>>>(xn, gamma_w, gamma_b, gamv);

  // scan
  k_init_state<<<(BB*DD*DD + 255)/256, 256, 0, stream>>>(M_init, S_init, Mg, Sg);
  const size_t smem = (size_t)SCAN_SMEM_FLOATS * sizeof(float);
  (void)hipFuncSetAttribute((const void*)k_scan, hipFuncAttributeMaxDynamicSharedMemorySize, (int)smem);
  k_scan<<<BB, 1024, smem, stream>>>(k_phi, vmb, gamv, q_phi, alpha, etab, theta,
                                     Mg, Sg, Xg, Tg, ysb);
  launch_gemm(ysb, memout_w, memout_b, nullptr, y_mem, BL, DD, DD, 0, stream);

  // MAG gate + output projection (+ residual)
  k_gate<<<BL, 256, 0, stream>>>(y_att, y_mem, ln_attn_w, ln_attn_b,
                                 ln_mem_w, ln_mem_b, gamma_at, gamma_me, gbuf);
  launch_gemm(gbuf, out_w, out_b, x, (float*)d_out, BL, DD, DD, 0, stream);
}